// RelGraphNet_10093173146053
// MI455X (gfx1250) — compile-verified
//
#include <hip/hip_runtime.h>

typedef __attribute__((ext_vector_type(2))) float v2f;
typedef __attribute__((ext_vector_type(8))) float v8f;

// ---------------------------------------------------------------- utilities
__global__ void fill_kernel(float* __restrict__ p, float v, int n) {
    int t = blockIdx.x * blockDim.x + threadIdx.x;
    if (t < n) p[t] = v;
}

// deg[dst] += 1 per edge (deg pre-seeded with 1.0 for the self-loop)
__global__ void deg_kernel(const int* __restrict__ dst, float* __restrict__ deg, int E) {
    int t = blockIdx.x * blockDim.x + threadIdx.x;
    if (t < E) unsafeAtomicAdd(&deg[dst[t]], 1.0f);
}

// in-place deg -> d^{-1/2}
__global__ void rsqrt_kernel(float* __restrict__ deg, int n) {
    int t = blockIdx.x * blockDim.x + threadIdx.x;
    if (t < n) {
        float d = deg[t];
        deg[t] = (d > 0.0f) ? rsqrtf(d) : 0.0f;
    }
}

// ---------------------------------------------------------------- WMMA GEMM
// C[M x NCOL] = A[M x K] * B[K x NCOL], fp32, one wave per 16x16 tile.
// Uses V_WMMA_F32_16X16X4_F32.  A frag (16x4): lanes 0-15 hold K={k,k+1},
// lanes 16-31 hold K={k+2,k+3}.  B frag (4x16) mirrored.  D: VGPR r holds
// rows (tm*16+r) [lanes 0-15] and (tm*16+8+r) [lanes 16-31].
template <int NCOL>
__global__ void gemm_wmma_kernel(const float* __restrict__ A,
                                 const float* __restrict__ B,
                                 float* __restrict__ C, int M, int K) {
    const int ntn     = NCOL / 16;
    const int mtiles  = (M + 15) / 16;
    const int total   = mtiles * ntn;
    int wave_g = blockIdx.x * (blockDim.x >> 5) + (threadIdx.x >> 5);
    if (wave_g >= total) return;                 // whole-wave uniform exit

    int tm   = wave_g / ntn;
    int tn   = wave_g % ntn;
    int lane = threadIdx.x & 31;
    int m    = lane & 15;                        // row within tile / col within tile
    int half = lane >> 4;                        // K-half select

    int arow = tm * 16 + m;
    if (arow >= M) arow = M - 1;                 // clamp (stores are guarded)
    const float* ap = A + (size_t)arow * K + half * 2;
    const float* bp = B + (size_t)(half * 2) * NCOL + tn * 16 + m;

    v8f acc = {};
#pragma unroll 4
    for (int k = 0; k < K; k += 4) {
        v2f a, b;
        a.x = ap[k];
        a.y = ap[k + 1];
        b.x = bp[(size_t)k * NCOL];
        b.y = bp[(size_t)(k + 1) * NCOL];
        acc = __builtin_amdgcn_wmma_f32_16x16x4_f32(
            /*neg_a=*/false, a, /*neg_b=*/false, b,
            /*c_mod=*/(short)0, acc, /*reuse_a=*/false, /*reuse_b=*/false);
    }

    int row0 = tm * 16 + half * 8;
    float* cp = C + (size_t)row0 * NCOL + tn * 16 + m;
#pragma unroll
    for (int r = 0; r < 8; ++r)
        if (row0 + r < M) cp[(size_t)r * NCOL] = acc[r];
}

// ---------------------------------------------------------------- scatter
// out[dst] += h[src] * dinv[src]*dinv[dst], C floats per edge, C/4 lanes/edge.
template <int C>
__global__ void scatter_kernel(const float* __restrict__ h,
                               const int* __restrict__ src,
                               const int* __restrict__ dst,
                               const float* __restrict__ dinv,
                               float* __restrict__ out, int E) {
    const int TPE = C / 4;
    long long t = (long long)blockIdx.x * blockDim.x + threadIdx.x;
    int e = (int)(t / TPE);
    if (e >= E) return;
    int j = (int)(t % TPE) * 4;

    int s = src[e], d = dst[e];
    float nrm = dinv[s] * dinv[d];
    float4 hv = *(const float4*)(h + (size_t)s * C + j);
    float* op = out + (size_t)d * C + j;
    unsafeAtomicAdd(op + 0, hv.x * nrm);
    unsafeAtomicAdd(op + 1, hv.y * nrm);
    unsafeAtomicAdd(op + 2, hv.z * nrm);
    unsafeAtomicAdd(op + 3, hv.w * nrm);
}

// ---------------------------------------------------------------- layer-1 epilogue
// o1 = relu(o1 + h1*dinv^2 + b1)   (self-loop term fused; race-free per node)
__global__ void finalize1_kernel(float* __restrict__ o1, const float* __restrict__ h1,
                                 const float* __restrict__ dinv,
                                 const float* __restrict__ b, int N) {
    int t = blockIdx.x * blockDim.x + threadIdx.x;
    if (t >= N * 64) return;
    int i = t >> 6, j = t & 63;
    float di = dinv[i];
    float v  = o1[t] + h1[t] * di * di + b[j];
    o1[t] = fmaxf(v, 0.0f);
}

// ---------------------------------------------------------------- layer-2 epilogue + mean-pool
// 256 threads = 8 nodes x 32 cols; LDS tree over the node dim, then one
// atomicAdd per column per block into pooled[32].
__global__ void finalize2_pool_kernel(const float* __restrict__ o2,
                                      const float* __restrict__ h2,
                                      const float* __restrict__ dinv,
                                      const float* __restrict__ b,
                                      float* __restrict__ pooled, int N) {
    __shared__ float sdata[256];
    int tid  = threadIdx.x;
    int node = blockIdx.x * 8 + (tid >> 5);
    int j    = tid & 31;

    float v = 0.0f;
    if (node < N) {
        float di = dinv[node];
        int   t  = node * 32 + j;
        v = fmaxf(o2[t] + h2[t] * di * di + b[j], 0.0f);
    }
    sdata[tid] = v;
    __syncthreads();
    if (tid < 128) sdata[tid] += sdata[tid + 128];
    __syncthreads();
    if (tid < 64) sdata[tid] += sdata[tid + 64];
    __syncthreads();
    if (tid < 32) unsafeAtomicAdd(&pooled[tid], sdata[tid] + sdata[tid + 32]);
}

// ---------------------------------------------------------------- final FC
__global__ void final_kernel(const float* __restrict__ pooled,
                             const float* __restrict__ fc_w,
                             const float* __restrict__ fc_b,
                             float* __restrict__ out, float invN) {
    if (threadIdx.x == 0 && blockIdx.x == 0) {
        float acc = 0.0f;
#pragma unroll
        for (int j = 0; j < 32; ++j) acc += pooled[j] * invN * fc_w[j];
        out[0] = acc + fc_b[0];
    }
}

// ---------------------------------------------------------------- launcher
extern "C" void kernel_launch(void* const* d_in, const int* in_sizes, int n_in,
                              void* d_out, int out_size, void* d_ws, size_t ws_size,
                              hipStream_t stream) {
    const float* x   = (const float*)d_in[0];
    const int*   ei  = (const int*)d_in[1];
    const float* W1  = (const float*)d_in[2];
    const float* b1  = (const float*)d_in[3];
    const float* W2  = (const float*)d_in[4];
    const float* b2  = (const float*)d_in[5];
    const float* fcw = (const float*)d_in[6];
    const float* fcb = (const float*)d_in[7];
    float*       out = (float*)d_out;

    const int IN = in_sizes[2] / 64;      // 256 (W1 is [256,64])
    const int N  = in_sizes[0] / IN;      // 100000
    const int E  = in_sizes[1] / 2;       // 3.2M
    const int* src = ei;
    const int* dst = ei + E;

    // workspace carve-out (floats, 64-element aligned)
    float* ws  = (float*)d_ws;
    size_t off = 0;
    auto carve = [&](size_t elems) {
        size_t o = off;
        off += (elems + 63) & ~(size_t)63;
        return ws + o;
    };
    float* deg    = carve((size_t)N);        // becomes dinv in-place
    float* h1     = carve((size_t)N * 64);
    float* o1     = carve((size_t)N * 64);
    float* h2     = carve((size_t)N * 32);
    float* o2     = carve((size_t)N * 32);
    float* pooled = carve(32);

    auto cdiv = [](long long a, long long b) { return (int)((a + b - 1) / b); };

    // degrees & symmetric normalization
    fill_kernel<<<cdiv(N, 256), 256, 0, stream>>>(deg, 1.0f, N);
    deg_kernel<<<cdiv(E, 256), 256, 0, stream>>>(dst, deg, E);
    rsqrt_kernel<<<cdiv(N, 256), 256, 0, stream>>>(deg, N);

    // layer 1: h1 = x @ W1 ; o1 = relu(A_hat h1 + b1)
    fill_kernel<<<cdiv((long long)N * 64, 256), 256, 0, stream>>>(o1, 0.0f, N * 64);
    {
        int tiles = ((N + 15) / 16) * (64 / 16);
        gemm_wmma_kernel<64><<<cdiv(tiles, 8), 256, 0, stream>>>(x, W1, h1, N, IN);
    }
    scatter_kernel<64><<<cdiv((long long)E * 16, 256), 256, 0, stream>>>(h1, src, dst, deg, o1, E);
    finalize1_kernel<<<cdiv((long long)N * 64, 256), 256, 0, stream>>>(o1, h1, deg, b1, N);

    // layer 2: h2 = o1 @ W2 ; pooled = sum relu(A_hat h2 + b2)
    fill_kernel<<<cdiv((long long)N * 32, 256), 256, 0, stream>>>(o2, 0.0f, N * 32);
    fill_kernel<<<1, 32, 0, stream>>>(pooled, 0.0f, 32);
    {
        int tiles = ((N + 15) / 16) * (32 / 16);
        gemm_wmma_kernel<32><<<cdiv(tiles, 8), 256, 0, stream>>>(o1, W2, h2, N, 64);
    }
    scatter_kernel<32><<<cdiv((long long)E * 8, 256), 256, 0, stream>>>(h2, src, dst, deg, o2, E);
    finalize2_pool_kernel<<<cdiv(N, 8), 256, 0, stream>>>(o2, h2, deg, b2, pooled, N);

    // mean + fc
    final_kernel<<<1, 32, 0, stream>>>(pooled, fcw, fcb, out, 1.0f / (float)N);
}